// TwoLevelSelectiveHardAttnLayer_73821897883845
// MI455X (gfx1250) — compile-verified
//
#include <hip/hip_runtime.h>
#include <hip/hip_bf16.h>
#include <stdint.h>

// Problem constants (fixed by the reference setup).
#define B_  8
#define T_  32768
#define D_  256
#define A_  128
#define H_  4
#define K1_ 1024
#define K2_ 512

typedef __attribute__((ext_vector_type(2))) float v2f;
typedef __attribute__((ext_vector_type(8))) float v8f;
typedef __attribute__((ext_vector_type(4))) int   v4i;

// Monotone float->uint key: larger float => larger uint.
__device__ __forceinline__ unsigned int f2ord(float f) {
    unsigned int u = __float_as_uint(f);
    return (u & 0x80000000u) ? ~u : (u | 0x80000000u);
}

// ---------------------------------------------------------------------------
// gfx1250 async global->LDS copy (ASYNCcnt path), with synchronous fallback.
// Builtin prototype (from clang diagnostic): (v4i AS1*, v4i AS3*, imm, imm).
// ---------------------------------------------------------------------------
#if defined(__AMDGCN__) && \
    __has_builtin(__builtin_amdgcn_global_load_async_to_lds_b128) && \
    __has_builtin(__builtin_amdgcn_s_wait_asynccnt)
#define USE_ASYNC_LDS 1
#else
#define USE_ASYNC_LDS 0
#endif

__device__ __forceinline__ void cp16_g2l(const float* g, float* l) {
#if USE_ASYNC_LDS
    __builtin_amdgcn_global_load_async_to_lds_b128(
        (__attribute__((address_space(1))) v4i*)
            (__attribute__((address_space(1))) const void*)g,
        (__attribute__((address_space(3))) v4i*)
            (__attribute__((address_space(3))) void*)l,
        0, 0);
#else
    *(float4*)l = *(const float4*)g;
#endif
}

__device__ __forceinline__ void lds_stage_fence() {
#if USE_ASYNC_LDS
    __builtin_amdgcn_s_wait_asynccnt(0);
#endif
    __syncthreads();
}

// ---------------------------------------------------------------------------
// Kernel 0: collapse the affine score maps.
//   wv[d]     = sum_a w[d,a] * v[a]          bv     = sum_a b[a] * v[a]
//   wvh[h,d]  = sum_a ws[h,d,a] * vs[h,a]    bvh[h] = sum_a bs[h,a] * vs[h,a]
// ---------------------------------------------------------------------------
__global__ __launch_bounds__(256)
void collapse_weights_kernel(const float* __restrict__ w,
                             const float* __restrict__ b,
                             const float* __restrict__ v,
                             const float* __restrict__ ws,
                             const float* __restrict__ bs,
                             const float* __restrict__ vs,
                             float* __restrict__ wv,
                             float* __restrict__ bv,
                             float* __restrict__ wvh,
                             float* __restrict__ bvh)
{
    const int d = threadIdx.x;  // 0..255
    float s = 0.f;
    for (int a = 0; a < A_; ++a) s += w[d * A_ + a] * v[a];
    wv[d] = s;
    for (int h = 0; h < H_; ++h) {
        float sh = 0.f;
        const float* wsh = ws + ((size_t)h * D_ + d) * A_;
        const float* vsh = vs + h * A_;
        for (int a = 0; a < A_; ++a) sh += wsh[a] * vsh[a];
        wvh[h * D_ + d] = sh;
    }
    if (d == 0) {
        float s0 = 0.f;
        for (int a = 0; a < A_; ++a) s0 += b[a] * v[a];
        bv[0] = s0;
        for (int h = 0; h < H_; ++h) {
            float sh = 0.f;
            for (int a = 0; a < A_; ++a) sh += bs[h * A_ + a] * vs[h * A_ + a];
            bvh[h] = sh;
        }
    }
}

// ---------------------------------------------------------------------------
// Kernel 1: level-0 scores. score0[row] = memory[row,:]·wv + bv
// One wave per row; 8 waves per block. This is the 268 MB bandwidth sweep.
// ---------------------------------------------------------------------------
__global__ __launch_bounds__(256)
void score0_kernel(const float* __restrict__ memory,
                   const float* __restrict__ wv,
                   const float* __restrict__ bv,
                   float* __restrict__ score0)
{
    __shared__ float s_wv[D_];
    s_wv[threadIdx.x] = wv[threadIdx.x];
    __syncthreads();

    const int wave = threadIdx.x >> 5;
    const int lane = threadIdx.x & 31;
    const size_t row = (size_t)blockIdx.x * 8 + wave;   // < B_*T_

    const float* r = memory + row * D_;
    // speculative prefetch of the next block's slice (gfx1250 global_prefetch)
    __builtin_prefetch(r + 8 * D_, 0, 0);

    float4 a0 = *(const float4*)(r + lane * 4);
    float4 a1 = *(const float4*)(r + 128 + lane * 4);
    float4 w0 = *(const float4*)(s_wv + lane * 4);
    float4 w1 = *(const float4*)(s_wv + 128 + lane * 4);
    float p = a0.x * w0.x + a0.y * w0.y + a0.z * w0.z + a0.w * w0.w
            + a1.x * w1.x + a1.y * w1.y + a1.z * w1.z + a1.w * w1.w;
    #pragma unroll
    for (int off = 16; off > 0; off >>= 1) p += __shfl_xor(p, off, 32);
    if (lane == 0) score0[row] = p + bv[0];
}

// ---------------------------------------------------------------------------
// Kernel 2: per-batch exact top-1024 (sorted descending, ties -> lower index).
// Composite 64-bit key (ordkey<<32)|~t is distinct per element, so an 8-pass
// MSB-first radix select finds the exact 1024th-largest key; compact + bitonic
// sort of the 1024 winners reproduces lax.top_k ordering.
// Dynamic LDS: keys[32768] u32 | sel[1024] u64 | hist[256] u32  (~137 KB)
// ---------------------------------------------------------------------------
__global__ __launch_bounds__(1024)
void topk1_kernel(const float* __restrict__ score0,
                  int* __restrict__ ind1)
{
    extern __shared__ unsigned char dynmem[];
    unsigned int*       keys = (unsigned int*)dynmem;                        // 32768
    unsigned long long* sel  = (unsigned long long*)(dynmem + 131072);       // 1024
    unsigned int*       hist = (unsigned int*)(dynmem + 131072 + 8192);      // 256

    __shared__ unsigned long long s_prefix;
    __shared__ int s_remaining;
    __shared__ int s_cnt;

    const int b   = blockIdx.x;
    const int tid = threadIdx.x;

    for (int t = tid; t < T_; t += 1024)
        keys[t] = f2ord(score0[(size_t)b * T_ + t]);
    if (tid == 0) { s_prefix = 0ull; s_remaining = K1_; s_cnt = 0; }
    __syncthreads();

    // MSB-first radix select, 8 bits per pass.
    for (int p = 7; p >= 0; --p) {
        for (int i = tid; i < 256; i += 1024) hist[i] = 0u;
        __syncthreads();
        const unsigned long long pref = s_prefix;
        for (int t = tid; t < T_; t += 1024) {
            unsigned long long c =
                ((unsigned long long)keys[t] << 32) | (unsigned int)(~t);
            bool match = (p == 7) ||
                         ((c >> ((p + 1) * 8)) == (pref >> ((p + 1) * 8)));
            if (match)
                atomicAdd(&hist[(unsigned int)(c >> (p * 8)) & 0xFFu], 1u);
        }
        __syncthreads();
        if (tid == 0) {
            int rem = s_remaining;
            unsigned long long np = s_prefix;
            for (int bin = 255; bin >= 0; --bin) {
                int cnt = (int)hist[bin];
                if (cnt < rem) { rem -= cnt; }
                else {
                    np |= ((unsigned long long)bin) << (p * 8);
                    break;
                }
            }
            s_prefix = np;
            s_remaining = rem;
        }
        __syncthreads();
    }

    // s_prefix == exact 1024th-largest composite key. Compact winners.
    const unsigned long long thr = s_prefix;
    for (int t = tid; t < T_; t += 1024) {
        unsigned long long c =
            ((unsigned long long)keys[t] << 32) | (unsigned int)(~t);
        if (c >= thr) {
            int pos = atomicAdd(&s_cnt, 1);
            if (pos < K1_) sel[pos] = c;
        }
    }
    __syncthreads();

    // Bitonic sort descending: 1024 elements, 1024 threads.
    for (unsigned int ksz = 2; ksz <= K1_; ksz <<= 1) {
        for (unsigned int j = ksz >> 1; j > 0; j >>= 1) {
            unsigned int i = tid, ixj = i ^ j;
            if (ixj > i) {
                unsigned long long a = sel[i], c = sel[ixj];
                bool up = ((i & ksz) == 0);
                if (up ? (a < c) : (a > c)) { sel[i] = c; sel[ixj] = a; }
            }
            __syncthreads();
        }
    }

    int t = (int)(~((unsigned int)(sel[tid] & 0xFFFFFFFFull)));
    ind1[b * K1_ + tid] = t;
}

// ---------------------------------------------------------------------------
// Kernel 3: level-1 scores on gathered rows, softmax (-> probs output),
// rank-based exact top-512 selection (set only; order irrelevant for max).
// One block per (b,h), 1024 threads = 32 waves; wave-per-row matvec.
// ---------------------------------------------------------------------------
__global__ __launch_bounds__(1024)
void level1_score_select_kernel(const float* __restrict__ memory,
                                const float* __restrict__ wvh,
                                const float* __restrict__ bvh,
                                const int* __restrict__ ind1,
                                float* __restrict__ probs_out,
                                int* __restrict__ sel_t)
{
    __shared__ float s_wvh[D_];
    __shared__ float s_sc[K1_];
    __shared__ unsigned long long s_key[K1_];
    __shared__ float s_red[32];
    __shared__ int s_cnt;

    const int bh   = blockIdx.x;       // b*H + h
    const int h    = bh & (H_ - 1);
    const int b    = bh >> 2;
    const int tid  = threadIdx.x;
    const int wave = tid >> 5;
    const int lane = tid & 31;

    if (tid < D_) s_wvh[tid] = wvh[h * D_ + tid];
    if (tid == 0) s_cnt = 0;
    __syncthreads();

    const float bias = bvh[h];
    for (int iter = 0; iter < 32; ++iter) {
        const int k = iter * 32 + wave;
        const int t = ind1[b * K1_ + k];
        const float* r = memory + ((size_t)b * T_ + t) * D_;
        float4 a0 = *(const float4*)(r + lane * 4);
        float4 a1 = *(const float4*)(r + 128 + lane * 4);
        float4 w0 = *(const float4*)(s_wvh + lane * 4);
        float4 w1 = *(const float4*)(s_wvh + 128 + lane * 4);
        float p = a0.x * w0.x + a0.y * w0.y + a0.z * w0.z + a0.w * w0.w
                + a1.x * w1.x + a1.y * w1.y + a1.z * w1.z + a1.w * w1.w;
        #pragma unroll
        for (int off = 16; off > 0; off >>= 1) p += __shfl_xor(p, off, 32);
        if (lane == 0) s_sc[k] = p + bias;
    }
    __syncthreads();

    // Block softmax over 1024 scores (one score per thread).
    float mx = s_sc[tid];
    #pragma unroll
    for (int off = 16; off > 0; off >>= 1) mx = fmaxf(mx, __shfl_xor(mx, off, 32));
    if (lane == 0) s_red[wave] = mx;
    __syncthreads();
    if (wave == 0) {
        float v = s_red[lane];
        #pragma unroll
        for (int off = 16; off > 0; off >>= 1) v = fmaxf(v, __shfl_xor(v, off, 32));
        if (lane == 0) s_red[0] = v;
    }
    __syncthreads();
    const float gmax = s_red[0];
    const float e = __expf(s_sc[tid] - gmax);
    __syncthreads();   // before reusing s_red
    float sm = e;
    #pragma unroll
    for (int off = 16; off > 0; off >>= 1) sm += __shfl_xor(sm, off, 32);
    if (lane == 0) s_red[wave] = sm;
    __syncthreads();
    if (wave == 0) {
        float v = s_red[lane];
        #pragma unroll
        for (int off = 16; off > 0; off >>= 1) v += __shfl_xor(v, off, 32);
        if (lane == 0) s_red[0] = v;
    }
    __syncthreads();
    const float prob = e / s_red[0];
    probs_out[(size_t)bh * K1_ + tid] = prob;

    // Exact top-512 set via rank counting on distinct composite keys.
    s_key[tid] = ((unsigned long long)f2ord(s_sc[tid]) << 32) | (unsigned int)(~tid);
    __syncthreads();
    const unsigned long long my = s_key[tid];
    int rank = 0;
    for (int j = 0; j < K1_; ++j) rank += (s_key[j] > my);
    if (rank < K2_) {
        int pos = atomicAdd(&s_cnt, 1);
        sel_t[bh * K2_ + pos] = ind1[b * K1_ + tid];
    }
}

// ---------------------------------------------------------------------------
// Kernel 4: lin1 on the selected 512 rows via V_WMMA_F32_16X16X4_F32.
// One block per (b,h); wave w owns N-tile [16w,16w+16); loops 32 M-tiles.
//
// Data movement (gfx1250): ws[h] rows 0-63 (64x128) staged once into LDS; each 16-row
// gathered A tile staged cooperatively into a double-buffered LDS tile via
// GLOBAL_LOAD_ASYNC_TO_LDS_B128 (ASYNCcnt + s_wait_asynccnt), overlapping the
// next tile's gather with WMMA on the current tile. Inner loop touches only
// LDS, so the matrix pipe is no longer gated on s_wait_loadcnt.
//
// LDS layout (floats), bank-conflict-free under the 16x16x4 lane mapping:
//   A tiles: 2 x [16 rows x stride 260]   (lanes 0-15 -> banks 4l, 16-31 -> 4l+2)
//   B:           [256 rows x stride 136]  (K and K+2 rows 16 banks apart)
// ---------------------------------------------------------------------------
#define A_STRIDE 260
#define B_STRIDE 136
#define A_TILE   (16 * A_STRIDE)                 // 4160 floats
#define LDS4_FLOATS (2 * A_TILE + D_ * B_STRIDE) // 8320 + 34816 = 43136
#define LDS4_BYTES  (LDS4_FLOATS * 4)            // 172544 B (<320 KB WGP LDS)

__global__ __launch_bounds__(256)
void lin1_wmma_max_kernel(const float* __restrict__ memory,
                          const float* __restrict__ ws,
                          const float* __restrict__ bs,
                          const float* __restrict__ vs,
                          const int* __restrict__ sel_t,
                          float* __restrict__ attns)
{
    extern __shared__ float lds[];
    float* A_lds = lds;                 // 2 * A_TILE
    float* B_lds = lds + 2 * A_TILE;    // D_ * B_STRIDE

    const int bh   = blockIdx.x;
    const int h    = bh & (H_ - 1);
    const int b    = bh >> 2;
    const int tid  = threadIdx.x;
    const int wave = tid >> 5;           // 0..7 -> N tile
    const int lane = tid & 31;
    const int lnn  = lane & 15;
    const int ksel = lane >> 4;          // 0: K{0,1}, 1: K{2,3}

    const float* ws_h  = ws + (size_t)h * D_ * A_;
    const float* mem_b = memory + (size_t)b * T_ * D_;
    const int n0 = wave * 16;
    const int n  = n0 + lnn;
    const int* sel = sel_t + bh * K2_;

    const float bias  = bs[h * A_ + n];
    const float scale = vs[h * A_ + n];

    // ---- stage B = ws[h] rows 0-63 (64x128) into LDS, padded stride ----
    #pragma unroll
    for (int jj = 0; jj < 8; ++jj) {                 // 2048 float4s / 256 thr
        const int i  = jj * 256 + tid;               // float4 index
        const int k  = (i * 4) >> 7;                 // /128
        const int nn = (i * 4) & 127;
        cp16_g2l(ws_h + i * 4, B_lds + k * B_STRIDE + nn);
    }

    // ---- stage gathered A tile m into buffer buf (16 rows x 256) ----
    auto stageA = [&](int buf, int m) {
        float* dst = A_lds + buf * A_TILE;
        #pragma unroll
        for (int jj = 0; jj < 4; ++jj) {             // 1024 float4s / 256 thr
            const int idx = jj * 256 + tid;
            const int r   = idx >> 6;                // row within tile
            const int c4  = idx & 63;                // float4 within row
            const int t   = sel[m * 16 + r];
            cp16_g2l(mem_b + (size_t)t * D_ + c4 * 4,
                     dst + r * A_STRIDE + c4 * 4);
        }
    };

    int buf = 0;
    stageA(0, 0);
    lds_stage_fence();   // covers B + first A tile

    float colmax = -3.402823466e38f;

    for (int m = 0; m < K2_ / 16; ++m) {
        if (m + 1 < K2_ / 16) stageA(buf ^ 1, m + 1);   // async overlap

        const float* Ab = A_lds + buf * A_TILE + lnn * A_STRIDE;
        v8f acc = {0.f, 0.f, 0.f, 0.f, 0.f, 0.f, 0.f, 0.f};
        #pragma unroll 8
        for (int kk = 0; kk < D_ / 4; ++kk) {
            const int k0 = kk * 4 + 2 * ksel;
            const float2 a2 = *(const float2*)(Ab + k0);
            v2f a;  a.x = a2.x;  a.y = a2.y;
            v2f bbv;
            bbv.x = B_lds[k0 * B_STRIDE + n];
            bbv.y = B_lds[(k0 + 1) * B_STRIDE + n];
            acc = __builtin_amdgcn_wmma_f32_16x16x4_f32(
                false, a, false, bbv, (short)0, acc, false, false);
        }
        #pragma unroll
        for (int j = 0; j < 8; ++j) {
            float val = (acc[j] + bias) * scale;
            colmax = fmaxf(colmax, val);
        }

        lds_stage_fence();   // next tile staged & everyone done with buf
        buf ^= 1;
    }

    // lanes L and L+16 hold the same n for complementary M rows
    colmax = fmaxf(colmax, __shfl_xor(colmax, 16, 32));
    if (lane < 16) attns[(size_t)bh * A_ + n] = colmax;
}

// ---------------------------------------------------------------------------
extern "C" void kernel_launch(void* const* d_in, const int* in_sizes, int n_in,
                              void* d_out, int out_size, void* d_ws, size_t ws_size,
                              hipStream_t stream) {
    const float* memory = (const float*)d_in[0];   // [B,T,D]
    const float* w      = (const float*)d_in[1];   // [D,A]
    const float* b      = (const float*)d_in[2];   // [A]
    const float* v      = (const float*)d_in[3];   // [A]
    const float* ws     = (const float*)d_in[4];   // [H,D,A]
    const float* bs     = (const float*)d_in[5];   // [H,A]
    const float* vs     = (const float*)d_in[6];   // [H,A]

    // Outputs, concatenated flat in return order: attns, probs, ind1.
    float* attns = (float*)d_out;                       // B*H*A   = 4096
    float* probs = attns + B_ * H_ * A_;                // B*H*K1  = 32768
    int*   ind1  = (int*)(probs + B_ * H_ * K1_);       // B*K1    = 8192

    // Workspace layout (floats).
    float* ws_f   = (float*)d_ws;
    float* wv     = ws_f;               // 256
    float* bv     = ws_f + 256;         // 1
    float* wvh    = ws_f + 260;         // H*D = 1024
    float* bvh    = ws_f + 1284;        // 4
    float* score0 = ws_f + 1536;        // B*T = 262144
    int*   sel_t  = (int*)(score0 + (size_t)B_ * T_);   // B*H*K2 = 16384 ints

    collapse_weights_kernel<<<1, 256, 0, stream>>>(w, b, v, ws, bs, vs,
                                                   wv, bv, wvh, bvh);

    score0_kernel<<<(B_ * T_) / 8, 256, 0, stream>>>(memory, wv, bv, score0);

    const size_t smem2 = 131072 /*keys*/ + 8192 /*sel*/ + 1024 /*hist*/;
    topk1_kernel<<<B_, 1024, smem2, stream>>>(score0, ind1);

    level1_score_select_kernel<<<B_ * H_, 1024, 0, stream>>>(
        memory, wvh, bvh, ind1, probs, sel_t);

    lin1_wmma_max_kernel<<<B_ * H_, 256, LDS4_BYTES, stream>>>(
        memory, ws, bs, vs, sel_t, attns);
}